// SingleStageDetector_19155554140841
// MI455X (gfx1250) — compile-verified
//
#include <hip/hip_runtime.h>

// ---- problem constants (from reference) ----
#define W_IMG 56
#define HW    3136        // 56*56
#define CIN   256
#define CHID  128
#define ANUM  9
#define NGT   40
#define BATCH 16
#define OUTCH 48          // conv2 channels computed (need 0..44; 3 tiles of 16)
#define PREDS (ANUM * HW) // 28224

typedef __attribute__((ext_vector_type(2))) float v2f;
typedef __attribute__((ext_vector_type(8))) float v8f;
typedef __attribute__((ext_vector_type(4))) int   v4i;

#define AS1 __attribute__((address_space(1)))
#define AS3 __attribute__((address_space(3)))

__device__ __forceinline__ void wait_asynccnt0() {
#if __has_builtin(__builtin_amdgcn_s_wait_asynccnt)
    __builtin_amdgcn_s_wait_asynccnt(0);
#else
    asm volatile("s_wait_asynccnt 0x0" ::: "memory");
#endif
}

// ============================================================================
// Kernel 1: fused conv1 (leaky relu) + conv2, fp32 WMMA 16x16x4.
// Block = 256 threads = 8 waves. Grid = (HW/16 pixel tiles, BATCH).
//   phase 0: stage X tile (256 x 16 f32, 16KB) into LDS — via the gfx1250
//            async global->LDS engine (no VGPR round trip) when available
//   phase 1: wave w computes hidden rows [16w, 16w+16) for this pixel tile
//            (K=256 -> 64 wmma), bias + leaky relu -> LDS Ht (128 x 16)
//   phase 2: waves 0..2 compute conv2 rows [16w, 16w+16) (K=128 -> 32 wmma),
//            bias, store raw channels to workspace.
// ============================================================================
__global__ __launch_bounds__(256)
void fused_conv_kernel(const float* __restrict__ F,
                       const float* __restrict__ W1,
                       const float* __restrict__ b1,
                       const float* __restrict__ W2,
                       const float* __restrict__ b2,
                       float* __restrict__ outbuf)
{
    __shared__ float Xt[CIN][16];   // 16 KB: feature tile (k, n)
    __shared__ float Ht[CHID][16];  //  8 KB: hidden tile  (m, n)

    const int n0  = blockIdx.x * 16;
    const int b   = blockIdx.y;
    const int tid = threadIdx.x;

    const float* __restrict__ Xb = F + (size_t)b * CIN * HW;

    // ---- phase 0: stage feature tile into LDS ----
#if __has_builtin(__builtin_amdgcn_global_load_async_to_lds_b128)
    {
        // 1024 16-byte quads; 4 async issues per thread, ASYNCcnt-tracked,
        // bypassing VGPRs entirely (GLOBAL_LOAD_ASYNC_TO_LDS_B128).
        for (int i = tid; i < CIN * 4; i += 256) {
            const int k = i >> 2;
            const int n = (i & 3) << 2;
            __builtin_amdgcn_global_load_async_to_lds_b128(
                (AS1 v4i*)(Xb + (size_t)k * HW + n0 + n),
                (AS3 v4i*)&Xt[k][n], 0, 0);
        }
        wait_asynccnt0();
    }
#else
    for (int i = tid; i < CIN * 16; i += 256) {
        const int k = i >> 4;
        const int n = i & 15;
        Xt[k][n] = Xb[(size_t)k * HW + n0 + n];
    }
#endif
    __syncthreads();

    const int wave = tid >> 5;
    const int lane = tid & 31;
    const int col  = lane & 15;              // N (or M for A-frag) index
    const int koff = (lane < 16) ? 0 : 2;    // K half per ISA f32 A/B layout
    const int mrb  = (lane < 16) ? 0 : 8;    // C/D row base per ISA layout

    // ---- phase 1: conv1 tile for this wave ----
    {
        const int m0 = wave * 16;
        const float* __restrict__ Arow = W1 + (size_t)(m0 + col) * CIN;
        v8f acc = {};
        for (int k = 0; k < CIN; k += 4) {
            v2f a, bf;
            a.x  = Arow[k + koff];
            a.y  = Arow[k + koff + 1];
            bf.x = Xt[k + koff][col];
            bf.y = Xt[k + koff + 1][col];
            acc = __builtin_amdgcn_wmma_f32_16x16x4_f32(
                      false, a, false, bf, (short)0, acc, false, false);
        }
#pragma unroll
        for (int v = 0; v < 8; ++v) {
            const int m = m0 + mrb + v;
            float val = acc[v] + b1[m];
            val = (val > 0.0f) ? val : 0.01f * val;   // leaky relu
            Ht[m][col] = val;
        }
    }
    __syncthreads();

    // ---- phase 2: conv2 (only channels 0..47; 45..47 are harmless extra) ----
    if (wave < 3) {
        const int m0 = wave * 16;
        const float* __restrict__ Arow = W2 + (size_t)(m0 + col) * CHID;
        v8f acc = {};
        for (int k = 0; k < CHID; k += 4) {
            v2f a, bf;
            a.x  = Arow[k + koff];
            a.y  = Arow[k + koff + 1];
            bf.x = Ht[k + koff][col];
            bf.y = Ht[k + koff + 1][col];
            acc = __builtin_amdgcn_wmma_f32_16x16x4_f32(
                      false, a, false, bf, (short)0, acc, false, false);
        }
        float* __restrict__ Ob = outbuf + (size_t)b * OUTCH * HW;
#pragma unroll
        for (int v = 0; v < 8; ++v) {
            const int m = m0 + mrb + v;
            Ob[(size_t)m * HW + n0 + col] = acc[v] + b2[m];
        }
    }
}

// ============================================================================
// Kernel 2: box decode + IoU matrix. Block = 256 threads handles 256
// predictions of one batch image. Output writes (the 72 MB stream that
// dominates this kernel) are fully coalesced: linear index i = pl*40 + n.
// ============================================================================
__global__ __launch_bounds__(256)
void decode_iou_kernel(const float* __restrict__ outbuf,
                       const float* __restrict__ anc,
                       const float* __restrict__ bboxes,
                       float* __restrict__ iou)
{
    __shared__ float g_xtl[NGT], g_ytl[NGT], g_xbr[NGT], g_ybr[NGT], g_area[NGT];
    __shared__ float p_xtl[256], p_ytl[256], p_xbr[256], p_ybr[256], p_area[256];

    const int b   = blockIdx.y;
    const int p0  = blockIdx.x * 256;
    const int tid = threadIdx.x;

    // stage ground-truth boxes (+ areas) for this batch image
    if (tid < NGT) {
        const float* gb = bboxes + ((size_t)b * NGT + tid) * 5;
        const float x0 = gb[0], y0 = gb[1], x1 = gb[2], y1 = gb[3];
        g_xtl[tid] = x0; g_ytl[tid] = y0;
        g_xbr[tid] = x1; g_ybr[tid] = y1;
        g_area[tid] = (x1 - x0) * (y1 - y0);
    }

    // decode this thread's prediction into LDS
    const int p = p0 + tid;
    if (p < PREDS) {
        const int a   = p / HW;
        const int pix = p - a * HW;
        const int py  = pix / W_IMG;
        const int px  = pix - py * W_IMG;

        const float* __restrict__ Ob = outbuf + (size_t)b * OUTCH * HW;
        const float t1 = Ob[(size_t)(5 * a + 1) * HW + pix];
        const float t2 = Ob[(size_t)(5 * a + 2) * HW + pix];
        const float tw = Ob[(size_t)(5 * a + 3) * HW + pix];
        const float th = Ob[(size_t)(5 * a + 4) * HW + pix];

        const float tx = 1.0f / (1.0f + __expf(-t1)) - 0.5f;   // sigmoid - 0.5
        const float ty = 1.0f / (1.0f + __expf(-t2)) - 0.5f;
        const float xc = (float)px + 0.5f + tx;
        const float yc = (float)py + 0.5f + ty;
        const float wp = anc[2 * a]     * __expf(tw);
        const float hp = anc[2 * a + 1] * __expf(th);

        const float xtl = xc - 0.5f * wp, xbr = xc + 0.5f * wp;
        const float ytl = yc - 0.5f * hp, ybr = yc + 0.5f * hp;
        p_xtl[tid] = xtl; p_ytl[tid] = ytl;
        p_xbr[tid] = xbr; p_ybr[tid] = ybr;
        p_area[tid] = (xbr - xtl) * (ybr - ytl);
    }
    __syncthreads();

    // coalesced write of the 256 x 40 IoU panel
    const int npred = min(256, PREDS - p0);
    const int total = npred * NGT;
    float* __restrict__ outb = iou + ((size_t)b * PREDS + p0) * NGT;
    for (int i = tid; i < total; i += 256) {
        const int pl = i / NGT;
        const int n  = i - pl * NGT;
        float iw = fminf(g_xbr[n], p_xbr[pl]) - fmaxf(g_xtl[n], p_xtl[pl]);
        float ih = fminf(g_ybr[n], p_ybr[pl]) - fmaxf(g_ytl[n], p_ytl[pl]);
        iw = fmaxf(iw, 0.0f);
        ih = fmaxf(ih, 0.0f);
        const float inter = iw * ih;
        outb[i] = inter / (p_area[pl] + g_area[n] - inter);
    }
}

// ============================================================================
extern "C" void kernel_launch(void* const* d_in, const int* in_sizes, int n_in,
                              void* d_out, int out_size, void* d_ws, size_t ws_size,
                              hipStream_t stream)
{
    const float* F      = (const float*)d_in[0]; // (16,256,56,56)
    const float* W1     = (const float*)d_in[1]; // (128,256)
    const float* b1     = (const float*)d_in[2]; // (128,)
    const float* W2     = (const float*)d_in[3]; // (65,128)
    const float* b2     = (const float*)d_in[4]; // (65,)
    const float* anc    = (const float*)d_in[5]; // (9,2)
    const float* bboxes = (const float*)d_in[6]; // (16,40,5)

    float* outbuf = (float*)d_ws;                // 16*48*3136 f32 = 9.6 MB

    dim3 g1(HW / 16, BATCH);                     // 196 x 16 blocks, 8 waves each
    fused_conv_kernel<<<g1, 256, 0, stream>>>(F, W1, b1, W2, b2, outbuf);

    dim3 g2((PREDS + 255) / 256, BATCH);         // 111 x 16 blocks
    decode_iou_kernel<<<g2, 256, 0, stream>>>(outbuf, anc, bboxes, (float*)d_out);
}